// SpatialCorrelationSampler_18253611008373
// MI455X (gfx1250) — compile-verified
//
#include <hip/hip_runtime.h>
#include <stdint.h>

// Spatial correlation (cost volume): B=8, C=256, H=64, W=96, patch 21x21,
// K=1x1, stride 1, pad 0, dilation 1.
//
// Per (b, i, di) the outputs form a |dj|<=10 band of the Gram matrix
// G[j,j'] = sum_c x1[c,j] * x2[c,j'].  Band computed as 16 WMMA 16x16 tiles
// (V_WMMA_F32_16X16X4_F32, K=256 -> 64 chained WMMAs, full fp32).
//
// CDNA5 data path:
//  - Rows staged transposed ([j][c]) into LDS by GLOBAL_LOAD_ASYNC_TO_LDS_B32
//    (per-lane global + LDS addresses = free transpose, ASYNCcnt tracked).
//  - x2 row double-buffered; s_wait_asynccnt sits at the TOP of the next
//    iteration, so staging overlaps compute + band store of the current one.
//  - Band buffer zeroed once up front; the WMMA scatter rewrites every
//    in-band entry each iteration (scatter and zero sets are disjoint), so
//    steady-state iterations do no zeroing and need only 2 barriers.

#define B_    8
#define C_    256
#define H_    64
#define W_    96
#define PATCH 21
#define RAD   10
#define KST   258             // LDS row stride (floats): even (8B v2f loads),
                              // 258 % 64 == 2 -> conflict-free lane banks
#define BSZ   (W_ * KST)      // 24768 floats per staged row
#define AOFF  0
#define B0OFF BSZ
#define B1OFF (2 * BSZ)
#define OOFF  (3 * BSZ)
#define OSZ   ((PATCH + 1) * W_)   // 21 band rows + 1 dump row
#define SMEMF (3 * BSZ + OSZ)      // 76416 floats = 305,664 bytes (<320KB)

typedef __attribute__((ext_vector_type(2))) float v2f;
typedef __attribute__((ext_vector_type(8))) float v8f;

// Async-stage one [C][W] global row into LDS transposed as [j][c].
// Coverage per thread: c = wave + 16q (q=0..15), j = lane + 32r (r=0..2).
// Global side: 32 consecutive j -> contiguous 128B per issue.
// LDS side: dword addr j*258 + c -> banks 2*lane + const, all 32 distinct.
// Uniform-stride address walk: two adds per issue, no division.
__device__ __forceinline__ void async_stage_row(uint32_t lds, uint32_t go,
                                                uint64_t gbase) {
#pragma unroll 1
  for (int q = 0; q < 16; ++q) {
#pragma unroll
    for (int r = 0; r < 3; ++r) {
      asm volatile("global_load_async_to_lds_b32 %0, %1, %2"
                   :: "v"(lds), "v"(go), "s"(gbase)
                   : "memory");
      lds += 4u * 32u * KST;          // j += 32
      go  += 4u * 32u;
    }
    lds += 4u * 16u;                  // c += 16 ...
    lds -= 4u * 96u * KST;            // ... j -= 96
    go  += 4u * 16u * (H_ * W_);
    go  -= 4u * 96u;
  }
}

__launch_bounds__(512, 1)
__global__ void corr_wmma_f32_async_kernel(const float* __restrict__ x1,
                                           const float* __restrict__ x2,
                                           float* __restrict__ out) {
  __shared__ __align__(16) float smem[SMEMF];

  const int i    = blockIdx.x;           // output row
  const int b    = blockIdx.y;           // batch
  const int tid  = (int)threadIdx.x;     // 0..511 (16 wave32s)
  const int lane = tid & 31;
  const int wave = tid >> 5;
  const int lm   = lane & 15;
  const int kh   = (lane >> 4) << 1;     // K offset 0 or 2 (f32 WMMA A/B layout)

  // tile pairs (jt, j2t): jt = (w+1)/3, j2t = jt + ((w+1)%3) - 1
  const int wp  = wave + 1;
  const int jt  = wp / 3;
  const int j2t = jt + (wp - 3 * jt) - 1;

  const uint32_t smem_base = (uint32_t)(uintptr_t)(void*)&smem[0];
  const uint64_t x1b = (uint64_t)(uintptr_t)(x1 + (size_t)b * C_ * H_ * W_);
  const uint64_t x2b = (uint64_t)(uintptr_t)(x2 + (size_t)b * C_ * H_ * W_);

  const uint32_t ldsA  = smem_base + 4u * (uint32_t)(AOFF  + lane * KST + wave);
  const uint32_t ldsB0 = smem_base + 4u * (uint32_t)(B0OFF + lane * KST + wave);
  const uint32_t ldsB1 = smem_base + 4u * (uint32_t)(B1OFF + lane * KST + wave);
  const uint32_t goLane = 4u * (uint32_t)(wave * (H_ * W_) + lane);

  // ---- prologue: async-stage x1 row i and (if valid) x2 row i-10 ----
  async_stage_row(ldsA, goLane + 4u * (uint32_t)(i * W_), x1b);
  if (i >= RAD)
    async_stage_row(ldsB0, goLane + 4u * (uint32_t)((i - RAD) * W_), x2b);

  // zero the whole band buffer once (covers j'-OOB entries for all pi)
  for (int idx = tid; idx < OSZ; idx += 512) smem[OOFF + idx] = 0.0f;

  // per-thread store offsets (pi-invariant): idx = tid + t*512 -> pj*6144 + j
  int soff[4];
#pragma unroll
  for (int t = 0; t < 4; ++t) {
    const int idx = tid + t * 512;
    const int pj  = idx / W_;
    soff[t] = pj * (H_ * W_) + (idx - pj * W_);
  }

  const int arow  = AOFF + (jt * 16 + lm) * KST + kh;   // lane = M row (x1 j)
  const int brow0 = (j2t * 16 + lm) * KST + kh;         // lane = N col (x2 j')
  float* const outb = out + (size_t)b * (PATCH * PATCH * H_ * W_)
                          + (size_t)i * W_;

  for (int pi = 0; pi < PATCH; ++pi) {
    const int i2 = i + pi - RAD;
    const bool valid = (i2 >= 0) && (i2 < H_);

    // current buffer's staging (issued last iteration) must be done,
    // and all waves must see it (plus prev Olds readers done)
    asm volatile("s_wait_asynccnt 0x0" ::: "memory");
    __syncthreads();

    // ---- issue async stage of the NEXT x2 row into the spare buffer ----
    const int i2n = i2 + 1;
    if (pi + 1 < PATCH && i2n >= 0 && i2n < H_)
      async_stage_row((pi & 1) ? ldsB0 : ldsB1,
                      goLane + 4u * (uint32_t)(i2n * W_), x2b);

    if (valid) {
      // ---- 16x16 Gram tile, K=256 via 64 chained V_WMMA_F32_16X16X4_F32 ----
      const int brow = brow0 + ((pi & 1) ? B1OFF : B0OFF);
      v8f acc = {0.f, 0.f, 0.f, 0.f, 0.f, 0.f, 0.f, 0.f};
#pragma unroll 8
      for (int k = 0; k < C_; k += 4) {
        const v2f a  = *(const v2f*)&smem[arow + k];
        const v2f bb = *(const v2f*)&smem[brow + k];
        acc = __builtin_amdgcn_wmma_f32_16x16x4_f32(
            false, a, false, bb, (short)0, acc, false, false);
      }

      // ---- branchless band scatter (out-of-band -> dump row 21) ----
      // C/D layout: VGPR r, lane l: M = r + 8*(l/16), N = l%16
      const int j2 = j2t * 16 + lm;
      const int j0 = jt * 16 + ((lane >> 4) << 3);
#pragma unroll
      for (int r = 0; r < 8; ++r) {
        const int j   = j0 + r;
        const int pj  = j2 - j + RAD;
        const int pjc = ((unsigned)pj < (unsigned)PATCH) ? pj : PATCH;
        smem[OOFF + pjc * W_ + j] = acc[r];
      }
    } else {
      // row-shift out of range: whole band is zero for this pi
#pragma unroll
      for (int t = 0; t < 4; ++t) {
        const int idx = tid + t * 512;
        if (idx < PATCH * W_) smem[OOFF + idx] = 0.0f;
      }
    }

    __syncthreads();

    // ---- coalesced store of the [21][96] band for this (b, pi, i) ----
    const float* src = &smem[OOFF];
    float* obase = outb + (size_t)pi * (PATCH * H_ * W_);
#pragma unroll
    for (int t = 0; t < 4; ++t) {
      const int idx = tid + t * 512;
      if (idx < PATCH * W_) obase[soff[t]] = src[idx];
    }
  }
}

extern "C" void kernel_launch(void* const* d_in, const int* in_sizes, int n_in,
                              void* d_out, int out_size, void* d_ws, size_t ws_size,
                              hipStream_t stream) {
  const float* x1 = (const float*)d_in[0];
  const float* x2 = (const float*)d_in[1];
  float* out = (float*)d_out;
  dim3 grid(H_, B_);  // 64 rows x 8 batches = 512 workgroups of 16 waves
  corr_wmma_f32_async_kernel<<<grid, dim3(512), 0, stream>>>(x1, x2, out);
}